// GNN_33689723470134
// MI455X (gfx1250) — compile-verified
//
#include <hip/hip_runtime.h>

typedef float v2f __attribute__((ext_vector_type(2)));
typedef float v8f __attribute__((ext_vector_type(8)));

#define N_NODES   50000
#define N_EDGES   800000
#define IN_CH     128
#define HID       96
#define N_CLASSES 10
#define N_GRAPHS  64
#define POOL_NPB  64   // nodes per pooling block

// ---------------------------------------------------------------------------
// Pass 0: graph-only stats. deg[i] = in-degree, sattr[i] = sum of edge_attr
// over incoming edges. Computed once, reused by both layers.
// ---------------------------------------------------------------------------
__global__ void edge_stats_kernel(const int* __restrict__ dst,
                                  const float* __restrict__ eattr,
                                  float* __restrict__ deg,
                                  float* __restrict__ sattr) {
  int e = blockIdx.x * blockDim.x + threadIdx.x;
  if (e >= N_EDGES) return;
  int d = dst[e];
  atomicAdd(&deg[d], 1.0f);
  atomicAdd(&sattr[d], eattr[e]);
}

// ---------------------------------------------------------------------------
// Dense node linear: out[N, HID] = X[N, K] @ W[K, HID] + bias, via
// V_WMMA_F32_16X16X4_F32 (fp32 parity with the reference).
// One wave32 per 16-node row tile; 6 accumulators cover HID = 6*16 columns.
// ---------------------------------------------------------------------------
template <int K>
__global__ void node_linear_wmma(const float* __restrict__ X,
                                 const float* __restrict__ W,     // [K, HID]
                                 const float* __restrict__ bias,  // [HID]
                                 float* __restrict__ out) {       // [N, HID]
  const int lane = threadIdx.x & 31;
  const int wave = blockIdx.x * (blockDim.x >> 5) + (threadIdx.x >> 5);
  const int row_base = wave * 16;
  if (row_base >= N_NODES) return;   // wave-uniform: EXEC stays all-ones for WMMA

  const int m     = lane & 15;        // A: row within tile; B/C: column within tile
  const int khalf = (lane >> 4) << 1; // lanes 0-15 carry K+0,K+1; lanes 16-31 carry K+2,K+3

  v8f acc[6];
#pragma unroll
  for (int t = 0; t < 6; ++t) acc[t] = {};

  const float* __restrict__ xrow = X + (size_t)(row_base + m) * K;

  for (int kk = 0; kk < K; kk += 4) {
    v2f a;
    a.x = xrow[kk + khalf + 0];       // A VGPR0: K=0 (lanes 0-15) / K=2 (lanes 16-31)
    a.y = xrow[kk + khalf + 1];       // A VGPR1: K=1 / K=3
#pragma unroll
    for (int t = 0; t < 6; ++t) {
      const int ncol = t * 16 + m;
      v2f b;
      b.x = W[(size_t)(kk + khalf + 0) * HID + ncol];
      b.y = W[(size_t)(kk + khalf + 1) * HID + ncol];
      acc[t] = __builtin_amdgcn_wmma_f32_16x16x4_f32(
          /*neg_a=*/false, a, /*neg_b=*/false, b,
          /*c_mod=*/(short)0, acc[t], /*reuse_a=*/false, /*reuse_b=*/false);
    }
  }

  // C/D layout: lane L -> column m; VGPR r -> row r + (L<16 ? 0 : 8)
  const int row_off = (lane < 16) ? 0 : 8;
#pragma unroll
  for (int t = 0; t < 6; ++t) {
    const int col = t * 16 + m;
    const float bv = bias[col];
#pragma unroll
    for (int r = 0; r < 8; ++r) {
      out[(size_t)(row_base + row_off + r) * HID + col] = acc[t][r] + bv;
    }
  }
}

// ---------------------------------------------------------------------------
// Edge scatter: agg[dst] += hlin[src], one wave per edge (96 = 3 floats/lane).
// Fire-and-forget fp32 atomics resolve in L2 (working set fits 192 MB L2).
// ---------------------------------------------------------------------------
__global__ void edge_scatter_kernel(const float* __restrict__ hlin,
                                    const int* __restrict__ src,
                                    const int* __restrict__ dst,
                                    float* __restrict__ agg) {
  const int wave = blockIdx.x * (blockDim.x >> 5) + (threadIdx.x >> 5);
  const int lane = threadIdx.x & 31;
  if (wave >= N_EDGES) return;
  const int s = src[wave];
  const int d = dst[wave];
  const float* __restrict__ hs = hlin + (size_t)s * HID;
  float* __restrict__ ad = agg + (size_t)d * HID;
  atomicAdd(&ad[lane +  0], hs[lane +  0]);
  atomicAdd(&ad[lane + 32], hs[lane + 32]);
  atomicAdd(&ad[lane + 64], hs[lane + 64]);
}

// ---------------------------------------------------------------------------
// h = relu(agg + sattr[i]*eW + deg[i]*eb)  (edge-MLP rank-1 correction folded in)
// ---------------------------------------------------------------------------
__global__ void finalize_relu_kernel(const float* __restrict__ agg,
                                     const float* __restrict__ deg,
                                     const float* __restrict__ sattr,
                                     const float* __restrict__ eW,
                                     const float* __restrict__ eb,
                                     float* __restrict__ h) {
  int idx = blockIdx.x * blockDim.x + threadIdx.x;
  if (idx >= N_NODES * HID) return;
  int i = idx / HID, c = idx - i * HID;
  float v = agg[idx] + sattr[i] * eW[c] + deg[i] * eb[c];
  h[idx] = v > 0.0f ? v : 0.0f;
}

// ---------------------------------------------------------------------------
// Mean pool: batch is sorted, so each block accumulates runs locally and only
// flushes an atomic on segment change (few atomics per block, low contention).
// ---------------------------------------------------------------------------
__global__ void pool_kernel(const float* __restrict__ h,
                            const int* __restrict__ batch,
                            float* __restrict__ gsum,   // [G, HID]
                            float* __restrict__ gcnt) { // [G]
  const int c = threadIdx.x;                 // 0..95
  const int start = blockIdx.x * POOL_NPB;
  if (start >= N_NODES) return;
  const int end = min(start + POOL_NPB, N_NODES);
  int cur = batch[start];
  float acc = 0.0f, cl = 0.0f;
  for (int i = start; i < end; ++i) {
    const int g = batch[i];
    if (g != cur) {
      atomicAdd(&gsum[(size_t)cur * HID + c], acc);
      if (c == 0) atomicAdd(&gcnt[cur], cl);
      acc = 0.0f; cl = 0.0f; cur = g;
    }
    acc += h[(size_t)i * HID + c];
    cl  += 1.0f;
  }
  atomicAdd(&gsum[(size_t)cur * HID + c], acc);
  if (c == 0) atomicAdd(&gcnt[cur], cl);
}

// ---------------------------------------------------------------------------
// out[G, 10] = (gsum / max(cnt,1)) @ Wout + bout   (tiny; plain VALU)
// ---------------------------------------------------------------------------
__global__ void out_kernel(const float* __restrict__ gsum,
                           const float* __restrict__ gcnt,
                           const float* __restrict__ Wout,  // [HID, 10]
                           const float* __restrict__ bout,
                           float* __restrict__ out) {
  int idx = blockIdx.x * blockDim.x + threadIdx.x;
  if (idx >= N_GRAPHS * N_CLASSES) return;
  int g = idx / N_CLASSES, o = idx - g * N_CLASSES;
  float inv = 1.0f / fmaxf(gcnt[g], 1.0f);
  float s = bout[o];
  for (int c = 0; c < HID; ++c)
    s += gsum[(size_t)g * HID + c] * inv * Wout[c * N_CLASSES + o];
  out[idx] = s;
}

// ---------------------------------------------------------------------------
extern "C" void kernel_launch(void* const* d_in, const int* in_sizes, int n_in,
                              void* d_out, int out_size, void* d_ws, size_t ws_size,
                              hipStream_t stream) {
  const float* x     = (const float*)d_in[0];
  const int*   eidx  = (const int*)  d_in[1];   // [2, E]: row0=src, row1=dst
  const float* eattr = (const float*)d_in[2];
  const int*   batch = (const int*)  d_in[3];
  const float* W1    = (const float*)d_in[4];
  const float* b1    = (const float*)d_in[5];
  const float* eW1   = (const float*)d_in[6];
  const float* eb1   = (const float*)d_in[7];
  const float* W2    = (const float*)d_in[8];
  const float* b2    = (const float*)d_in[9];
  const float* eW2   = (const float*)d_in[10];
  const float* eb2   = (const float*)d_in[11];
  const float* Wout  = (const float*)d_in[12];
  const float* bout  = (const float*)d_in[13];
  float* out = (float*)d_out;

  const int* src = eidx;
  const int* dst = eidx + N_EDGES;

  // workspace layout (floats)
  float* ws    = (float*)d_ws;
  float* deg   = ws;                                  // [N]
  float* sattr = deg   + N_NODES;                     // [N]
  float* gsum  = sattr + N_NODES;                     // [G*HID]
  float* gcnt  = gsum  + (size_t)N_GRAPHS * HID;      // [G]
  float* hlin  = gcnt  + N_GRAPHS;                    // [N*HID]
  float* agg   = hlin  + (size_t)N_NODES * HID;       // [N*HID]
  float* h     = agg   + (size_t)N_NODES * HID;       // [N*HID]

  const size_t nh_bytes = (size_t)N_NODES * HID * sizeof(float);

  // zero deg, sattr, gsum, gcnt (contiguous prefix)
  hipMemsetAsync(ws, 0,
                 (size_t)(2 * N_NODES + N_GRAPHS * HID + N_GRAPHS) * sizeof(float),
                 stream);

  edge_stats_kernel<<<(N_EDGES + 255) / 256, 256, 0, stream>>>(dst, eattr, deg, sattr);

  const int gemm_blocks = (N_NODES / 16 + 7) / 8;     // 8 waves (16-row tiles) per block

  // ---- layer 1 ----
  node_linear_wmma<IN_CH><<<gemm_blocks, 256, 0, stream>>>(x, W1, b1, hlin);
  hipMemsetAsync(agg, 0, nh_bytes, stream);
  edge_scatter_kernel<<<N_EDGES / 8, 256, 0, stream>>>(hlin, src, dst, agg);
  finalize_relu_kernel<<<(N_NODES * HID + 255) / 256, 256, 0, stream>>>(
      agg, deg, sattr, eW1, eb1, h);

  // ---- layer 2 ----
  node_linear_wmma<HID><<<gemm_blocks, 256, 0, stream>>>(h, W2, b2, hlin);
  hipMemsetAsync(agg, 0, nh_bytes, stream);
  edge_scatter_kernel<<<N_EDGES / 8, 256, 0, stream>>>(hlin, src, dst, agg);
  finalize_relu_kernel<<<(N_NODES * HID + 255) / 256, 256, 0, stream>>>(
      agg, deg, sattr, eW2, eb2, h);

  // ---- pooling + output head ----
  pool_kernel<<<(N_NODES + POOL_NPB - 1) / POOL_NPB, HID, 0, stream>>>(h, batch, gsum, gcnt);
  out_kernel<<<1, N_GRAPHS * N_CLASSES, 0, stream>>>(gsum, gcnt, Wout, bout, out);
}